// MAFM_9929964388462
// MI455X (gfx1250) — compile-verified
//
#include <hip/hip_runtime.h>
#include <math.h>

// Problem constants from the reference.
#define IMG_D 8192
#define IND_D 192
#define TOT_D (IMG_D + IND_D)   // 8384
#define B_ROWS 128
#define NCLS 3

typedef __attribute__((ext_vector_type(2))) float v2f;
typedef __attribute__((ext_vector_type(8))) float v8f;

// The entire reference network collapses (layernorm over a size-1 axis == its
// bias), so the exact output is:
//   logits[j] = c1 * sum(lin_w[j, 0:8192]) + c2 * sum(lin_w[j, 8192:8384]) + lin_b[j]
//   out[b, :] = softmax(logits) for every b.
// We compute the row sums as lin_w @ ones with V_WMMA_F32_16X16X4_F32
// (exact fp32 MACs). One wave per class row j; 3 waves total.
__global__ __launch_bounds__(96)
void mafm_collapsed_kernel(const float* __restrict__ lin_w,
                           const float* __restrict__ lin_b,
                           const float* __restrict__ c1p,
                           const float* __restrict__ c2p,
                           float* __restrict__ out,
                           int out_size) {
    const int tid  = threadIdx.x;
    const int wave = tid >> 5;          // 0..2 -> class row j
    const int lane = tid & 31;

    // A-matrix (16x4 f32) lane layout: lane L holds A[L%16][2*(L/16) + {0,1}].
    // With A[m][k] = row[tile*64 + m*4 + k], lane L loads row[tile*64 + moff .. +1].
    const int moff = (lane & 15) * 4 + ((lane >> 4) << 1);  // 8-byte aligned

    const v2f bones = {1.0f, 1.0f};     // B all-ones: D[m][n] = sum_k A[m][k]

    const float* row = lin_w + wave * TOT_D;

    // ---- sum over d in [0, 8192): 128 tiles of 64, 4 accumulators to break the
    // WMMA->WMMA RAW chain on C.
    v8f acc0 = {}, acc1 = {}, acc2 = {}, acc3 = {};
    for (int t = 0; t < IMG_D / 64; t += 4) {
        v2f a0 = *(const v2f*)(row + (t + 0) * 64 + moff);
        v2f a1 = *(const v2f*)(row + (t + 1) * 64 + moff);
        v2f a2 = *(const v2f*)(row + (t + 2) * 64 + moff);
        v2f a3 = *(const v2f*)(row + (t + 3) * 64 + moff);
        acc0 = __builtin_amdgcn_wmma_f32_16x16x4_f32(false, a0, false, bones,
                                                     (short)0, acc0, false, false);
        acc1 = __builtin_amdgcn_wmma_f32_16x16x4_f32(false, a1, false, bones,
                                                     (short)0, acc1, false, false);
        acc2 = __builtin_amdgcn_wmma_f32_16x16x4_f32(false, a2, false, bones,
                                                     (short)0, acc2, false, false);
        acc3 = __builtin_amdgcn_wmma_f32_16x16x4_f32(false, a3, false, bones,
                                                     (short)0, acc3, false, false);
    }
    v8f accA = (acc0 + acc1) + (acc2 + acc3);

    // ---- sum over d in [8192, 8384): 3 tiles of 64.
    v8f accB = {};
    for (int t = 0; t < IND_D / 64; ++t) {
        v2f a = *(const v2f*)(row + IMG_D + t * 64 + moff);
        accB = __builtin_amdgcn_wmma_f32_16x16x4_f32(false, a, false, bones,
                                                     (short)0, accB, false, false);
    }

    // C/D layout (16x16 f32): lane 0 holds D[0..7][0] in its 8 regs, lane 16
    // holds D[8..15][0]. Horizontal-sum each lane's 8 regs, then combine
    // lanes 0 and 16 to get the full row sum (all columns are identical).
    float h1 = accA[0] + accA[1] + accA[2] + accA[3] +
               accA[4] + accA[5] + accA[6] + accA[7];
    float h2 = accB[0] + accB[1] + accB[2] + accB[3] +
               accB[4] + accB[5] + accB[6] + accB[7];
    float s1 = __shfl(h1, 0, 32) + __shfl(h1, 16, 32);
    float s2 = __shfl(h2, 0, 32) + __shfl(h2, 16, 32);

    const float c1 = *c1p;
    const float c2 = *c2p;

    __shared__ float sm_logits[NCLS];
    if (lane == 0) sm_logits[wave] = c1 * s1 + c2 * s2 + lin_b[wave];
    __syncthreads();

    // 3-way softmax (identical in every thread), then broadcast to all rows.
    float l0 = sm_logits[0], l1 = sm_logits[1], l2 = sm_logits[2];
    float mx = fmaxf(l0, fmaxf(l1, l2));
    float e0 = expf(l0 - mx), e1 = expf(l1 - mx), e2 = expf(l2 - mx);
    float inv = 1.0f / (e0 + e1 + e2);
    float p[NCLS] = {e0 * inv, e1 * inv, e2 * inv};

    for (int i = tid; i < out_size; i += blockDim.x) {
        out[i] = p[i % NCLS];
    }
}

extern "C" void kernel_launch(void* const* d_in, const int* in_sizes, int n_in,
                              void* d_out, int out_size, void* d_ws, size_t ws_size,
                              hipStream_t stream) {
    (void)in_sizes; (void)d_ws; (void)ws_size;

    // setup_inputs() order: image_embeds, indicator_embeds, l1{12 arrays},
    // l2{12 arrays}, lin_w, lin_b.  Handle both "nested dict fully flattened
    // to leaves" (n_in == 28) and "each top-level value flat" (n_in == 6).
    const float* c1p;   // l1.ln2_b
    const float* c2p;   // l2.ln2_b
    const float* lw;
    const float* lb;
    if (n_in >= 28) {
        // 0:image 1:indicator 2..13:l1 leaves 14..25:l2 leaves 26:lin_w 27:lin_b
        c1p = (const float*)d_in[13];   // l1.ln2_b (12th leaf of l1)
        c2p = (const float*)d_in[25];   // l2.ln2_b
        lw  = (const float*)d_in[26];
        lb  = (const float*)d_in[27];
    } else {
        // l1/l2 flattened to 16 floats each:
        // in_w[3] in_b[3] out_w out_b ln1_w ln1_b ff1_w ff1_b ff2_w ff2_b ln2_w ln2_b
        c1p = (const float*)d_in[2] + 15;
        c2p = (const float*)d_in[3] + 15;
        lw  = (const float*)d_in[4];
        lb  = (const float*)d_in[5];
    }

    mafm_collapsed_kernel<<<1, 96, 0, stream>>>(lw, lb, c1p, c2p,
                                                (float*)d_out, out_size);
}